// KMeansCompressor_69965017252468
// MI455X (gfx1250) — compile-verified
//
#include <hip/hip_runtime.h>
#include <hip/hip_bf16.h>
#include <stdint.h>

// Nearest-scalar-codebook quantizer: out[i] = argmin_k |x[i] - centers[k]|,
// N = 1<<20, K = 256, int32 indices out.
//
// MI455X reasoning:
//   - Traffic floor: 4MB in + 4MB out @ 23.3 TB/s ~ 0.35us.
//   - Brute force is N*K ~ 268M distance+select steps (~0.8G VALU ops) -> far
//     above the memory floor. A WMMA formulation (scores c^2 - 2xc) would
//     still need 268M MACs + 268M cross-lane compares and fp16/bf16 mantissas
//     would flip near-tie argmins; matrix cores are the wrong tool here.
//   - Optimal: sort the codebook once, then each element binary-searches the
//     255 midpoints: 8 probes/element. Levels 0-2 probe lane-UNIFORM
//     addresses (7 values total) -> resolved from SGPRs with v_cndmask;
//     levels 3-7 + the perm gather (6 probes) come from LDS.
//   - CDNA5-specific path: lookup tables staged global->LDS with
//     global_load_async_to_lds_b32, fenced by s_wait_asynccnt (ASYNCcnt).

#define K_CENTERS 256
#define ELEMS_PER_THREAD 8
#define BLOCK_THREADS 256

// ---------------------------------------------------------------------------
// Kernel 1: bitonic sort of (center value, original index), emit midpoints.
// ---------------------------------------------------------------------------
__global__ void __launch_bounds__(K_CENTERS)
kq_sort_centers(const float* __restrict__ centers,
                float* __restrict__ mid_out,   // 256 floats (255 used)
                int* __restrict__ perm_out) {  // 256 ints
    __shared__ float v[K_CENTERS];
    __shared__ int   p[K_CENTERS];
    const int t = threadIdx.x;

    v[t] = centers[t];
    p[t] = t;
    __syncthreads();

    // Ascending bitonic sort, stable via original-index tie-break so duplicate
    // center values keep first-occurrence order (matches argmin's first-index).
    for (int k = 2; k <= K_CENTERS; k <<= 1) {
        for (int j = k >> 1; j > 0; j >>= 1) {
            const int ixj = t ^ j;
            if (ixj > t) {
                const float a = v[t], b = v[ixj];
                const int pa = p[t], pb = p[ixj];
                const bool up = ((t & k) == 0);
                const bool gt = (a > b) || (a == b && pa > pb);
                if (up ? gt : !gt) {
                    v[t] = b; v[ixj] = a;
                    p[t] = pb; p[ixj] = pa;
                }
            }
            __syncthreads();
        }
    }

    // Decision boundaries: x goes left of mid[i] when x <= mid[i].
    if (t < K_CENTERS - 1) {
        mid_out[t] = 0.5f * (v[t] + v[t + 1]);
    } else {
        mid_out[K_CENTERS - 1] = __builtin_inff();  // sentinel, never probed
    }
    perm_out[t] = p[t];
}

// ---------------------------------------------------------------------------
// Kernel 2: vectorized binary-search quantizer.
// ---------------------------------------------------------------------------
__global__ void __launch_bounds__(BLOCK_THREADS)
kq_quantize(const float* __restrict__ x,
            const float* __restrict__ mid_g,
            const int* __restrict__ perm_g,
            int* __restrict__ out,
            int n) {
    __shared__ float mid[K_CENTERS];
    __shared__ int   perm[K_CENTERS];

    const int t = threadIdx.x;

    // --- CDNA5 async global->LDS staging of the lookup tables -------------
    // 256 threads each pull one midpoint + one perm entry straight into LDS
    // (no VGPR round trip). Each wave fences its own ASYNCcnt, then the
    // workgroup barrier publishes LDS to all waves.
    {
        const unsigned lds_m = (unsigned)(uintptr_t)&mid[t];   // low 32 bits = LDS offset
        const unsigned lds_p = (unsigned)(uintptr_t)&perm[t];
        const unsigned goff  = (unsigned)t * 4u;
        asm volatile("global_load_async_to_lds_b32 %0, %1, %2"
                     :: "v"(lds_m), "v"(goff), "s"(mid_g) : "memory");
        asm volatile("global_load_async_to_lds_b32 %0, %1, %2"
                     :: "v"(lds_p), "v"(goff), "s"(perm_g) : "memory");
    }

    // Lane-uniform probes for search levels 0-2 (7 scalars, SGPR-resident;
    // these s_loads overlap the async staging above).
    const float m127 = mid_g[127];
    const float m63  = mid_g[63];
    const float m191 = mid_g[191];
    const float m31  = mid_g[31];
    const float m95  = mid_g[95];
    const float m159 = mid_g[159];
    const float m223 = mid_g[223];

    asm volatile("s_wait_asynccnt 0x0" ::: "memory");
    __syncthreads();

    const int base = (blockIdx.x * BLOCK_THREADS + t) * ELEMS_PER_THREAD;

    if (base + ELEMS_PER_THREAD <= n) {
        // 32-byte aligned: two b128 loads.
        const float4 a = *(const float4*)(x + base);
        const float4 b = *(const float4*)(x + base + 4);
        float xs[ELEMS_PER_THREAD] = {a.x, a.y, a.z, a.w, b.x, b.y, b.z, b.w};

        int idx[ELEMS_PER_THREAD];

        // Levels 0-2: pure VALU (v_cmp + v_cndmask) against SGPR scalars.
        // Strict '>' sends exact-midpoint ties to the left (smaller) center.
#pragma unroll
        for (int e = 0; e < ELEMS_PER_THREAD; ++e) {
            const float xv = xs[e];
            const bool c0 = xv > m127;                       // step 128
            const float l1 = c0 ? m191 : m63;
            const bool c1 = xv > l1;                         // step 64
            const float l2 = c0 ? (c1 ? m223 : m159)
                                : (c1 ? m95  : m31);
            const bool c2 = xv > l2;                         // step 32
            idx[e] = (c0 ? 128 : 0) + (c1 ? 64 : 0) + (c2 ? 32 : 0);
        }

        // Levels 3-7: LDS probes; the 8 elements are independent dependency
        // chains, so each level issues 8 ds_loads back-to-back (latency
        // hidden without extra waves).
#pragma unroll
        for (int step = 16; step >= 1; step >>= 1) {
#pragma unroll
            for (int e = 0; e < ELEMS_PER_THREAD; ++e) {
                idx[e] += (xs[e] > mid[idx[e] + step - 1]) ? step : 0;
            }
        }

        int4 r0, r1;
        r0.x = perm[idx[0]]; r0.y = perm[idx[1]];
        r0.z = perm[idx[2]]; r0.w = perm[idx[3]];
        r1.x = perm[idx[4]]; r1.y = perm[idx[5]];
        r1.z = perm[idx[6]]; r1.w = perm[idx[7]];
        *(int4*)(out + base)     = r0;
        *(int4*)(out + base + 4) = r1;
    } else {
        // Tail guard (N = 1<<20 divides evenly, but stay safe).
        for (int i = base; i < n; ++i) {
            const float xv = x[i];
            int idx = 0;
#pragma unroll
            for (int step = 128; step >= 1; step >>= 1) {
                idx += (xv > mid[idx + step - 1]) ? step : 0;
            }
            out[i] = perm[idx];
        }
    }
}

// ---------------------------------------------------------------------------
// Host-side launcher (graph-capture safe: stream-ordered kernels only).
// ---------------------------------------------------------------------------
extern "C" void kernel_launch(void* const* d_in, const int* in_sizes, int n_in,
                              void* d_out, int out_size, void* d_ws, size_t ws_size,
                              hipStream_t stream) {
    const float* x       = (const float*)d_in[0];   // (N,) f32
    const float* centers = (const float*)d_in[1];   // (K,) f32
    int*         out     = (int*)d_out;             // (N,) int32 indices
    const int n = in_sizes[0];

    // Workspace layout: [0,1KB) midpoints f32[256], [1KB,2KB) perm i32[256].
    float* ws_mid  = (float*)d_ws;
    int*   ws_perm = (int*)((char*)d_ws + K_CENTERS * sizeof(float));

    kq_sort_centers<<<1, K_CENTERS, 0, stream>>>(centers, ws_mid, ws_perm);

    const int elems_per_block = BLOCK_THREADS * ELEMS_PER_THREAD;
    const int grid = (n + elems_per_block - 1) / elems_per_block;   // 512 for N=1<<20
    kq_quantize<<<grid, BLOCK_THREADS, 0, stream>>>(x, ws_mid, ws_perm, out, n);
}